// YoloLoss_13967233647276
// MI455X (gfx1250) — compile-verified
//
#include <hip/hip_runtime.h>

// Problem constants fixed by the reference.
#define NBATCH   16
#define NANCHOR  3
#define NROW     52
#define NCOL     52
#define MASK_N   (NBATCH * NANCHOR * NROW * NCOL)   // 129792 floats per mask
#define IGNORE_THRES 0.5f

// ---------------------------------------------------------------------------
// Kernel 1: obj_mask = 0, noobj_mask = 1. float4 stores -> global_store_b128.
// ---------------------------------------------------------------------------
__global__ void yolo_init_masks(float* __restrict__ obj,
                                float* __restrict__ noobj,
                                int n4) {
    int i = blockIdx.x * blockDim.x + threadIdx.x;
    if (i < n4) {
        float4 z = make_float4(0.0f, 0.0f, 0.0f, 0.0f);
        float4 o = make_float4(1.0f, 1.0f, 1.0f, 1.0f);
        reinterpret_cast<float4*>(obj)[i]   = z;
        reinterpret_cast<float4*>(noobj)[i] = o;
    }
}

// ---------------------------------------------------------------------------
// Kernel 2: per-target anchor assignment + ignore-threshold scatter.
// One thread per target. All scatter writes are constant values (1.0 to obj,
// 0.0 to noobj), so write races between targets hitting the same cell are
// benign and the result is deterministic.
// ---------------------------------------------------------------------------
__global__ void yolo_assign_targets(const float* __restrict__ target,   // [nT,6]
                                    const float* __restrict__ anchors,  // [6]
                                    float* __restrict__ obj,
                                    float* __restrict__ noobj,
                                    float* __restrict__ best_ious,
                                    int nT) {
    int i = blockIdx.x * blockDim.x + threadIdx.x;
    if (i >= nT) return;

    const float* t = target + i * 6;
    __builtin_prefetch(t, 0, 0);   // global_prefetch_b8 (tiny table, pulls whole line)

    // image_id = tile(arange(B), nT/B)[i] == i % B
    int   img   = i % NBATCH;
    float cls   = t[1];
    bool  valid = cls > -1.0f;

    // grid_scale = [nRow, nCol]
    float tx = t[2] * (float)NROW;
    float ty = t[3] * (float)NCOL;
    float tw = t[4] * (float)NROW;
    float th = t[5] * (float)NCOL;

    // .astype(int32): truncation toward zero (values are positive in-range)
    int gx = (int)tx;
    int gy = (int)ty;

    // IoU against each scaled anchor; strict '>' so ties pick the first
    // anchor, matching jnp.argmax. FP op order matches the reference:
    // union = ((aw*ah + 1e-16) + tw*th) - inter
    float iou[NANCHOR];
    float best_iou = -1.0f;
    int   best_n   = 0;
    float twth = tw * th;
#pragma unroll
    for (int a = 0; a < NANCHOR; ++a) {
        float aw = anchors[2 * a]     * (float)NROW;
        float ah = anchors[2 * a + 1] * (float)NCOL;
        float inter = fminf(aw, tw) * fminf(ah, th);
        float uni   = ((aw * ah + 1e-16f) + twth) - inter;
        iou[a] = inter / uni;
        if (iou[a] > best_iou) { best_iou = iou[a]; best_n = a; }
    }

    // best_ious is returned for ALL targets (no validity masking in reference)
    best_ious[i] = best_iou;

    // JAX scatter mode='drop': invalid targets or OOB cells are dropped.
    bool inb = (gx >= 0) && (gx < NROW) && (gy >= 0) && (gy < NCOL);
    if (valid && inb) {
        int cell = gx * NCOL + gy;
        int base = ((img * NANCHOR + best_n) * NROW * NCOL) + cell;
        obj[base]   = 1.0f;   // obj_mask[img, best_n, gx, gy] = 1
        noobj[base] = 0.0f;   // noobj_mask[img, best_n, gx, gy] = 0

        // ignore-threshold pass: any anchor with IoU > 0.5 clears noobj
#pragma unroll
        for (int a = 0; a < NANCHOR; ++a) {
            if (iou[a] > IGNORE_THRES) {
                noobj[((img * NANCHOR + a) * NROW * NCOL) + cell] = 0.0f;
            }
        }
    }
}

// ---------------------------------------------------------------------------
// Launch. Inputs (setup_inputs order): d_in[0]=x (UNUSED, shape-only in the
// reference), d_in[1]=target [nT,6] f32, d_in[2]=anchors [6] f32.
// Output layout (flat, return order): obj_mask | noobj_mask | best_ious.
// ---------------------------------------------------------------------------
extern "C" void kernel_launch(void* const* d_in, const int* in_sizes, int n_in,
                              void* d_out, int out_size, void* d_ws, size_t ws_size,
                              hipStream_t stream) {
    (void)n_in; (void)out_size; (void)d_ws; (void)ws_size;

    const float* target  = (const float*)d_in[1];
    const float* anchors = (const float*)d_in[2];

    float* out       = (float*)d_out;
    float* obj       = out;                 // MASK_N floats
    float* noobj     = out + MASK_N;        // MASK_N floats
    float* best_ious = out + 2 * MASK_N;    // nT floats

    int nT = in_sizes[1] / 6;               // 320
    int n4 = MASK_N / 4;                    // 32448 float4 stores per mask

    // Init is ~1 MB of stores: memory path is global_store_b128, fully
    // parallel across the device; stream order serializes it before scatter.
    yolo_init_masks<<<(n4 + 255) / 256, 256, 0, stream>>>(obj, noobj, n4);

    // 320 targets, 64-thread (2-wave) blocks.
    yolo_assign_targets<<<(nT + 63) / 64, 64, 0, stream>>>(
        target, anchors, obj, noobj, best_ious, nT);
}